// MedianBlur_62929860821123
// MI455X (gfx1250) — compile-verified
//
#include <hip/hip_runtime.h>

#define IMG_H 384
#define IMG_W 384
#define TILE_W 32
#define TILE_H 8
#define PAD 4                       // 9x9 kernel -> halo of 4
#define LDS_W (TILE_W + 2 * PAD)    // 40 floats = 160B (16B aligned rows)
#define LDS_H (TILE_H + 2 * PAD)    // 16
#define LDS_N (LDS_W * LDS_H)       // 640 floats
#define NGROUPS (LDS_N / 4)         // 160 float4 groups
#define GROUPS_PER_ROW (LDS_W / 4)  // 10
#define NTHREADS (TILE_W * TILE_H)  // 256 = 8 waves (wave32)

#if defined(__HIP_DEVICE_COMPILE__)
typedef int v4i_t __attribute__((vector_size(4 * sizeof(int))));
typedef __attribute__((address_space(1))) v4i_t gv4i_t;  // global 4xi32
typedef __attribute__((address_space(3))) v4i_t lv4i_t;  // LDS 4xi32
#endif

__device__ __forceinline__ void cswap(float& a, float& b) {
    float lo = fminf(a, b);
    float hi = fmaxf(a, b);
    a = lo;
    b = hi;
}

__global__ __launch_bounds__(NTHREADS)
void MedianBlur9_kernel(const float* __restrict__ img, float* __restrict__ out) {
    __shared__ __align__(16) float tile[LDS_N];

    const int tid   = threadIdx.y * TILE_W + threadIdx.x;
    const int plane = blockIdx.z;                       // b*C + c
    const float* src = img + (size_t)plane * (IMG_H * IMG_W);

    const int gx0 = blockIdx.x * TILE_W;                // output tile origin
    const int gy0 = blockIdx.y * TILE_H;

    // ---- Cooperative tile+halo load, float4 granularity ----
    // gx0-4 is a multiple of 4 and IMG_W%4==0, so each 4-float group is either
    // fully inside the image or fully outside -> zero padding stays exact.
    if (tid < NGROUPS) {
        const int r  = tid / GROUPS_PER_ROW;
        const int c4 = (tid - r * GROUPS_PER_ROW) * 4;
        const int gx = gx0 - PAD + c4;                  // group start x (mult of 4)
        const int gy = gy0 - PAD + r;
        if ((unsigned)gx < (unsigned)IMG_W && (unsigned)gy < (unsigned)IMG_H) {
#if defined(__HIP_DEVICE_COMPILE__) && __has_builtin(__builtin_amdgcn_global_load_async_to_lds_b128)
            __builtin_amdgcn_global_load_async_to_lds_b128(
                (gv4i_t*)(unsigned long long)(src + gy * IMG_W + gx),
                (lv4i_t*)&tile[r * LDS_W + c4],
                /*imm offset*/ 0, /*cpol*/ 0);
#else
            *(float4*)&tile[r * LDS_W + c4] =
                *(const float4*)(src + gy * IMG_W + gx);
#endif
        } else {
            float4 z; z.x = 0.f; z.y = 0.f; z.z = 0.f; z.w = 0.f;
            *(float4*)&tile[r * LDS_W + c4] = z;        // zero padding (ds_store_b128)
        }
    }

#if defined(__HIP_DEVICE_COMPILE__) && __has_builtin(__builtin_amdgcn_global_load_async_to_lds_b128)
#if __has_builtin(__builtin_amdgcn_s_wait_asynccnt)
    __builtin_amdgcn_s_wait_asynccnt(0);
#else
    asm volatile("s_wait_asynccnt 0" ::: "memory");
#endif
#endif
    __syncthreads();

    // ---- Forgetful selection: exact median of 81 with a 42-register buffer ----
    const float* wbase = &tile[threadIdx.y * LDS_W + threadIdx.x];

    float buf[42];
#pragma unroll
    for (int n = 0; n < 42; ++n) {
        buf[n] = wbase[(n / 9) * LDS_W + (n % 9)];
    }

#pragma unroll
    for (int n = 42; n < 81; ++n) {
        const int sz = 84 - n;  // current buffer size: 42 ... 4
        // forward pass: max -> buf[sz-1]
#pragma unroll
        for (int i = 0; i < sz - 1; ++i) cswap(buf[i], buf[i + 1]);
        // backward pass (excluding tail): min -> buf[0]
#pragma unroll
        for (int i = sz - 2; i >= 1; --i) cswap(buf[i - 1], buf[i]);
        // discard min (buf[0]) and max (buf[sz-1]); insert next window element
        buf[0] = wbase[(n / 9) * LDS_W + (n % 9)];
    }

    // buffer holds 3 candidates; median of 3 is the answer
    cswap(buf[0], buf[1]);
    cswap(buf[1], buf[2]);
    cswap(buf[0], buf[1]);

    const int ox = gx0 + threadIdx.x;
    const int oy = gy0 + threadIdx.y;
    out[(size_t)plane * (IMG_H * IMG_W) + (size_t)oy * IMG_W + ox] = buf[1];
}

extern "C" void kernel_launch(void* const* d_in, const int* in_sizes, int n_in,
                              void* d_out, int out_size, void* d_ws, size_t ws_size,
                              hipStream_t stream) {
    (void)in_sizes; (void)n_in; (void)d_ws; (void)ws_size; (void)out_size;
    const float* img = (const float*)d_in[0];
    float* out = (float*)d_out;

    dim3 block(TILE_W, TILE_H, 1);
    dim3 grid(IMG_W / TILE_W, IMG_H / TILE_H, 8 * 3);  // 12 x 48 x 24 blocks
    MedianBlur9_kernel<<<grid, block, 0, stream>>>(img, out);
}